// EPGSimulationTimeVaryingGradients_90142773608460
// MI455X (gfx1250) — compile-verified
//
#include <hip/hip_runtime.h>
#include <stdint.h>

// ---------------- problem constants ----------------
#define N_PULSES 64
#define BATCH    16384
#define N_STATES 21

#define LANES_PER_VOX 4
#define VOX_PER_WAVE  8            // 32 lanes / 4 lanes-per-voxel
#define WAVES_PER_BLOCK 8
#define BLOCK 256

#define TILE_DW        (VOX_PER_WAVE * N_STATES)     // 168 dwords per plane row
#define PLANE_STRIDE_DW ((uint32_t)BATCH * N_STATES) // 344064 dwords between planes
#define PULSE_STRIDE_BYTES ((uint64_t)5 * BATCH * N_STATES * 4) // 6,881,280 B

typedef unsigned int u32x4 __attribute__((ext_vector_type(4)));
typedef int          i32x4 __attribute__((ext_vector_type(4)));
typedef int          i32x8 __attribute__((ext_vector_type(8)));

// The TDM builtin only exists in the device pass; the host pass still parses
// __global__ bodies, so hide it behind __HIP_DEVICE_COMPILE__.
#if defined(__HIP_DEVICE_COMPILE__)
  #if !__has_builtin(__builtin_amdgcn_tensor_store_from_lds)
    #error "gfx1250 TDM builtin __builtin_amdgcn_tensor_store_from_lds not available in device pass"
  #endif
  #if defined(__clang_major__) && (__clang_major__ >= 23)
    #define TDM_STORE(g0, g1, g2, g3, z8) \
        __builtin_amdgcn_tensor_store_from_lds((g0), (g1), (g2), (g3), (z8), 0)
  #else
    #define TDM_STORE(g0, g1, g2, g3, z8) \
        __builtin_amdgcn_tensor_store_from_lds((g0), (g1), (g2), (g3), 0)
  #endif
#else
  #define TDM_STORE(g0, g1, g2, g3, z8) ((void)0)
#endif

__device__ __forceinline__ uint32_t rfl_u32(uint32_t v) {
    return (uint32_t)__builtin_amdgcn_readfirstlane((int)v);
}

__global__ __launch_bounds__(BLOCK)
void epg_tdm_kernel(const float* __restrict__ flip,
                    const float* __restrict__ phase,
                    const float* __restrict__ T1a,
                    const float* __restrict__ T2a,
                    const float* __restrict__ B0a,
                    const float* __restrict__ B1a,
                    const int*   __restrict__ TRp,
                    float* __restrict__ out)
{
    // per-wave double-buffered output tile: [wave][buf][plane(5)][vox(8)][state(21)]
    __shared__ float smem[WAVES_PER_BLOCK * 2 * 5 * TILE_DW]; // 53,760 B

    const int tid  = threadIdx.x;
    const int lane = tid & 31;
    const int r    = tid & 3;                 // sub-lane within voxel group
    const int vv   = lane >> 2;               // voxel index within wave (0..7)
    const int waveInBlock = tid >> 5;
    const int vox  = (blockIdx.x * BLOCK + tid) >> 2;

    // ---- per-voxel constants (hoisted out of the pulse loop) ----
    const float TRf = (float)TRp[0];
    const float t1 = T1a[vox], t2 = T2a[vox], b0 = B0a[vox], b1 = B1a[vox];
    const float E1 = expf(-TRf / t1);
    const float E2 = expf(-TRf / t2);
    const float onemE1 = 1.0f - E1;
    const float phi = 6.28318530717958647692f * b0 * TRf * 0.001f;
    float cphi, sphi; sincosf(phi, &sphi, &cphi);
    const float c2x = E2 * cphi;              // E2 * e^{+i phi} (Fp); conj used for Fm
    const float c2y = E2 * sphi;

    // ---- state carry: this lane owns states [S0, S0+NS) ----
    // r==0 -> 0..5 (6 states), r==1 -> 6..10, r==2 -> 11..15, r==3 -> 16..20
    const int S0 = (r == 0) ? 0 : (5 * r + 1);
    float Fpx[6], Fpy[6], Fmx[6], Fmy[6], Zs[6];
#pragma unroll
    for (int s = 0; s < 6; ++s) { Fpx[s]=0.f; Fpy[s]=0.f; Fmx[s]=0.f; Fmy[s]=0.f; Zs[s]=0.f; }
    if (r == 0) Zs[0] = 1.0f;                 // Z[0] = 1

    // ---- TDM descriptor constants ----
    const uint32_t voxBase = ((uint32_t)blockIdx.x * WAVES_PER_BLOCK + waveInBlock) * VOX_PER_WAVE;
    uint64_t gaddr = (uint64_t)(uintptr_t)out + (uint64_t)voxBase * (N_STATES * 4u);
    uint32_t lds_wave_bytes = (uint32_t)(uintptr_t)(&smem[waveInBlock * (2 * 5 * TILE_DW)]);
    lds_wave_bytes = rfl_u32(lds_wave_bytes);

    // D# group1 (static): data_size=4B, tensor_dim0=344064, tensor_dim1=64,
    // tile_dim0=168, tile_dim1=5, tensor_dim0_stride=344064
    const i32x8 g1 = { (int)0x00020000,                              // data_size=2 (4B)
                       (int)((PLANE_STRIDE_DW & 0xFFFFu) << 16),     // tensor_dim0 lo16
                       (int)((PLANE_STRIDE_DW >> 16) | (64u << 16)), // tdim0 hi16 | tdim1 lo16
                       (int)(((uint32_t)TILE_DW) << 16),             // tdim1 hi16(0) | tile_dim0
                       (int)5,                                       // tile_dim1=5 | tile_dim2=0
                       (int)PLANE_STRIDE_DW,                         // tensor_dim0_stride lo32
                       0, 0 };
    const i32x4 z4 = { 0, 0, 0, 0 };
    const i32x8 z8 = { 0, 0, 0, 0, 0, 0, 0, 0 };
    (void)z8;

    // ---------------- pulse recurrence ----------------
    for (int p = 0; p < N_PULSES; ++p) {
        const int buf = p & 1;

        // per-pulse uniform RF params
        const float a = flip[p];
        const float b = phase[p];
        float cb, sb;  sincosf(b, &sb, &cb);
        const float c2b = cb * cb - sb * sb;   // cos 2b
        const float s2b = 2.0f * cb * sb;      // sin 2b
        float ca, sa;  sincosf(0.5f * a * b1, &sa, &ca);
        const float ca2 = ca * ca, sa2 = sa * sa, cs = ca * sa, dZ = ca2 - sa2;

        float prevx = 0.f, prevy = 0.f;        // delayed Fp_n (shift up)
        float botmx = 0.f, botmy = 0.f;        // Fm_n at local s=0 (shift down)

#pragma unroll
        for (int s = 0; s < 6; ++s) {
            if (s < 5 || r == 0) {
                // relaxation + B0 precession (fused)
                const float ox = Fpx[s], oy = Fpy[s], mx = Fmx[s], my = Fmy[s];
                const float fpx = c2x * ox - c2y * oy;
                const float fpy = c2x * oy + c2y * ox;
                const float fmx = c2x * mx + c2y * my;   // * conj
                const float fmy = c2x * my - c2y * mx;
                const float inj = (s == 0 && r == 0) ? onemE1 : 0.f;
                const float z   = fmaf(E1, Zs[s], inj);

                // RF rotation
                const float t1x = fmx * c2b + fmy * s2b;       // conj(Fm)*e^{2ib}
                const float t1y = fmx * s2b - fmy * c2b;
                const float zcs = cs * z;
                const float fpnx = ca2 * fpx + sa2 * t1x - zcs * sb;
                const float fpny = ca2 * fpy + sa2 * t1y + zcs * cb;
                const float t2x = fpx * c2b - fpy * s2b;       // conj(Fp)*e^{-2ib}
                const float t2y = -(fpx * s2b + fpy * c2b);
                const float fmnx = sa2 * t2x + ca2 * fmx - zcs * sb;
                const float fmny = sa2 * t2y + ca2 * fmy - zcs * cb;
                const float zn = cs * (fpy * cb - fpx * sb - fmx * sb - fmy * cb) + dZ * z;

                Zs[s] = zn;
                if (s == 0) { botmx = fmnx; botmy = fmny; }       // goes to lane r-1 top
                else        { Fmx[s-1] = fmnx; Fmy[s-1] = fmny; } // Fm shift down
                if (s > 0)  { Fpx[s] = prevx; Fpy[s] = prevy; }   // Fp shift up (delayed)
                prevx = fpnx; prevy = fpny;
            }
        }

        // cross-lane boundary exchange within the 4-lane voxel group
        const float incx = __shfl_up(prevx, 1, 4);
        const float incy = __shfl_up(prevy, 1, 4);
        Fpx[0] = (r == 0) ? 0.f : incx;        // Fp_s[0] = 0 at global boundary
        Fpy[0] = (r == 0) ? 0.f : incy;
        float imx = __shfl_down(botmx, 1, 4);
        float imy = __shfl_down(botmy, 1, 4);
        imx = (r == 3) ? 0.f : imx;            // Fm_s[20] = 0 at global boundary
        imy = (r == 3) ? 0.f : imy;
        if (r == 0) { Fmx[5] = imx; Fmy[5] = imy; }
        else        { Fmx[4] = imx; Fmy[4] = imy; }

        // ---- stage this pulse's output tile in LDS (double buffered) ----
        // Make sure the TDM store issued 2 pulses ago (same buffer) has drained.
        __builtin_amdgcn_s_wait_tensorcnt(1);
        float* wbuf = &smem[waveInBlock * (2 * 5 * TILE_DW) + buf * (5 * TILE_DW)];
#pragma unroll
        for (int s = 0; s < 6; ++s) {
            if (s < 5 || r == 0) {
                const int o = vv * N_STATES + S0 + s;
                wbuf[0 * TILE_DW + o] = Fpx[s];
                wbuf[1 * TILE_DW + o] = Fpy[s];
                wbuf[2 * TILE_DW + o] = Fmx[s];
                wbuf[3 * TILE_DW + o] = Fmy[s];
                wbuf[4 * TILE_DW + o] = Zs[s];
            }
        }

        // LDS writes must land before the TDM reads them.
        asm volatile("s_wait_dscnt 0" ::: "memory");

        // ---- one async TDM store per wave per pulse: LDS tile -> global ----
        {
            const uint32_t glo = rfl_u32((uint32_t)(gaddr & 0xffffffffu));
            const uint32_t ghi = rfl_u32((uint32_t)(gaddr >> 32)) & 0x01ffffffu;
            u32x4 g0;
            g0.x = 1u;                                   // count=1, flags=0
            g0.y = lds_wave_bytes + (uint32_t)buf * (5u * TILE_DW * 4u);
            g0.z = glo;                                  // global_addr[31:0]
            g0.w = ghi | (2u << 30);                     // global_addr[56:32] | type=2
            TDM_STORE(g0, g1, z4, z4, z8);
        }

        gaddr += PULSE_STRIDE_BYTES;
    }
    // s_endpgm performs an implicit wait-idle (drains TENSORcnt).
}

extern "C" void kernel_launch(void* const* d_in, const int* in_sizes, int n_in,
                              void* d_out, int out_size, void* d_ws, size_t ws_size,
                              hipStream_t stream) {
    const float* flip  = (const float*)d_in[0];
    const float* phase = (const float*)d_in[1];
    const float* T1    = (const float*)d_in[2];
    const float* T2    = (const float*)d_in[3];
    const float* B0    = (const float*)d_in[4];
    const float* B1    = (const float*)d_in[5];
    const int*   TR    = (const int*)d_in[6];
    float* out = (float*)d_out;
    (void)in_sizes; (void)n_in; (void)out_size; (void)d_ws; (void)ws_size;

    const int totalThreads = BATCH * LANES_PER_VOX;       // 65536 -> 2048 wave32s
    epg_tdm_kernel<<<dim3(totalThreads / BLOCK), dim3(BLOCK), 0, stream>>>(
        flip, phase, T1, T2, B0, B1, TR, out);
}